// DEBIAS_BPR_60644938219975
// MI455X (gfx1250) — compile-verified
//
#include <hip/hip_runtime.h>
#include <hip/hip_bf16.h>

typedef __attribute__((ext_vector_type(2))) float v2f;
typedef __attribute__((ext_vector_type(8))) float v8f;

#define NN 4096          // 2*BATCH
#define EPS_D 1e-8f

// ws layout in floats:
//   [WS_R,  WS_R+32768)  r values   [2 losses][4096 rows][4 factors]  (float4 per row)
//   [WS_M,  WS_M+65536)  M matrix   [16 rows][4096]  rows l*8+0..3 = R_k, l*8+4..7 = C_k
//   [WS_S,  WS_S+14)     S pair sums [2][7]  order (00,01,11,12,22,23,33)
//   [WS_T,  WS_T+8)      T grand sums [2][4]
#define WS_R 0
#define WS_M 32768
#define WS_S 98304
#define WS_T 98318
#define WS_ACC_COUNT (WS_T + 8 - WS_M)   // region that must be zeroed each launch

// ---------------------------------------------------------------- kernel A: r
__global__ void k_rows(const float* __restrict__ u, const float* __restrict__ p,
                       const float* __restrict__ up, const float* __restrict__ pp,
                       float* __restrict__ ws) {
    int t = blockIdx.x * blockDim.x + threadIdx.x;   // 0..8191
    if (t >= 2 * NN) return;
    int loss = t >> 12;
    int i    = t & (NN - 1);
    const float* a = (loss == 0) ? u  : p;
    const float* b = (loss == 0) ? up : pp;
    const float* row = (i < 2048) ? (a + (size_t)i * 64)
                                  : (b + (size_t)(i - 2048) * 64);
    float4 out;
    float* o = (float*)&out;
#pragma unroll
    for (int k = 0; k < 4; ++k) {
        float4 v = *(const float4*)(row + 4 * k);    // columns 4k..4k+3 (only cols 0..15 used)
        o[k] = v.x * v.x + v.y * v.y + v.z * v.z + v.w * v.w;
    }
    ((float4*)(ws + WS_R))[(size_t)loss * NN + i] = out;
}

// ---------------------------------------------------------------- kernel Z: zero accumulators
__global__ void k_zero(float* __restrict__ ws) {
    int t = blockIdx.x * blockDim.x + threadIdx.x;
    if (t < WS_ACC_COUNT) ws[WS_M + t] = 0.0f;
}

// ---------------------------------------------------------------- wave reduce (wave32)
__device__ __forceinline__ float wave_red(float v) {
#pragma unroll
    for (int o = 16; o > 0; o >>= 1) v += __shfl_down(v, o, 32);
    return v;
}

// ---------------------------------------------------------------- kernel B: fused O(N^2) pass
// grid = (8 j-chunks, 16 row-blocks, 2 losses), block = 256
__global__ void k_main(const float* __restrict__ wsr, float* __restrict__ wsM,
                       float* __restrict__ wsS, float* __restrict__ wsT) {
    __shared__ float4 sr[NN];                         // 64 KB of the 320 KB WGP LDS
    const int loss = blockIdx.z;
    const int tid  = threadIdx.x;

    const float4* rsrc = ((const float4*)(wsr + WS_R)) + (size_t)loss * NN;
    for (int n = tid; n < NN; n += 256) sr[n] = rsrc[n];
    __syncthreads();

    const int i = blockIdx.y * 256 + tid;             // this thread's row
    const float4 ri = sr[i];

    float R0 = 0, R1 = 0, R2 = 0, R3 = 0;             // row sums of D_k
    float C0 = 0, C1 = 0, C2 = 0, C3 = 0;             // col sums of D_k (transposed orientation)
    float s00 = 0, s01 = 0, s11 = 0, s12 = 0, s22 = 0, s23 = 0, s33 = 0;

    const int j0 = blockIdx.x * 512;
#pragma unroll 4
    for (int j = j0; j < j0 + 512; ++j) {
        float4 rj = sr[j];                            // same address across wave -> LDS broadcast
        float t0 = rj.x - ri.x, t1 = rj.y - ri.y, t2 = rj.z - ri.z, t3 = rj.w - ri.w;
        float d0 = sqrtf(fmaxf(t0, 0.f) + EPS_D);
        float d1 = sqrtf(fmaxf(t1, 0.f) + EPS_D);
        float d2 = sqrtf(fmaxf(t2, 0.f) + EPS_D);
        float d3 = sqrtf(fmaxf(t3, 0.f) + EPS_D);
        float e0 = sqrtf(fmaxf(-t0, 0.f) + EPS_D);
        float e1 = sqrtf(fmaxf(-t1, 0.f) + EPS_D);
        float e2 = sqrtf(fmaxf(-t2, 0.f) + EPS_D);
        float e3 = sqrtf(fmaxf(-t3, 0.f) + EPS_D);
        R0 += d0; R1 += d1; R2 += d2; R3 += d3;
        C0 += e0; C1 += e1; C2 += e2; C3 += e3;
        s00 += d0 * d0; s01 += d0 * d1; s11 += d1 * d1;
        s12 += d1 * d2; s22 += d2 * d2; s23 += d2 * d3; s33 += d3 * d3;
    }

    // marginal accumulation: rows l*8+0..3 = R, l*8+4..7 = C
    const int rb = loss * 8;
    atomicAdd(&wsM[WS_M + (size_t)(rb + 0) * NN + i], R0);
    atomicAdd(&wsM[WS_M + (size_t)(rb + 1) * NN + i], R1);
    atomicAdd(&wsM[WS_M + (size_t)(rb + 2) * NN + i], R2);
    atomicAdd(&wsM[WS_M + (size_t)(rb + 3) * NN + i], R3);
    atomicAdd(&wsM[WS_M + (size_t)(rb + 4) * NN + i], C0);
    atomicAdd(&wsM[WS_M + (size_t)(rb + 5) * NN + i], C1);
    atomicAdd(&wsM[WS_M + (size_t)(rb + 6) * NN + i], C2);
    atomicAdd(&wsM[WS_M + (size_t)(rb + 7) * NN + i], C3);

    // wave-level reduce, then a few atomics per wave
    float tT0 = wave_red(R0), tT1 = wave_red(R1), tT2 = wave_red(R2), tT3 = wave_red(R3);
    float w00 = wave_red(s00), w01 = wave_red(s01), w11 = wave_red(s11), w12 = wave_red(s12);
    float w22 = wave_red(s22), w23 = wave_red(s23), w33 = wave_red(s33);
    if ((tid & 31) == 0) {
        atomicAdd(&wsT[WS_T + loss * 4 + 0], tT0);
        atomicAdd(&wsT[WS_T + loss * 4 + 1], tT1);
        atomicAdd(&wsT[WS_T + loss * 4 + 2], tT2);
        atomicAdd(&wsT[WS_T + loss * 4 + 3], tT3);
        atomicAdd(&wsS[WS_S + loss * 7 + 0], w00);
        atomicAdd(&wsS[WS_S + loss * 7 + 1], w01);
        atomicAdd(&wsS[WS_S + loss * 7 + 2], w11);
        atomicAdd(&wsS[WS_S + loss * 7 + 3], w12);
        atomicAdd(&wsS[WS_S + loss * 7 + 4], w22);
        atomicAdd(&wsS[WS_S + loss * 7 + 5], w23);
        atomicAdd(&wsS[WS_S + loss * 7 + 6], w33);
    }
}

// ---------------------------------------------------------------- kernel C: WMMA Gram + finish
// block = 256 (8 waves). G = M * M^T  (16 x 16, K = 4096) via chained V_WMMA_F32_16X16X4_F32.
// For a Gram, the ISA 16x4 A layout and 4x16 B layout coincide per lane:
//   lane L<16  holds M[L][k+0], M[L][k+1]; lane L>=16 holds M[L-16][k+2], M[L-16][k+3]
// so the SAME v2f register pair serves as both A and B operands.
__global__ void k_final(const float* __restrict__ ws, float* __restrict__ out) {
    __shared__ float sG[8 * 256 + 256];               // per-wave tiles + final G
    const int tid  = threadIdx.x;
    const int wave = tid >> 5;
    const int lane = tid & 31;
    const int m    = lane & 15;
    const int koff = (lane >> 4) << 1;                // 0 for lanes 0-15, 2 for lanes 16-31

    const float* rowp = ws + WS_M + (size_t)m * NN + koff;
    v8f acc = {0.f, 0.f, 0.f, 0.f, 0.f, 0.f, 0.f, 0.f};

    const int kb0 = wave * 512;
    for (int kb = kb0; kb < kb0 + 512; kb += 4) {
        v2f a;
        a.x = rowp[kb];
        a.y = rowp[kb + 1];
        acc = __builtin_amdgcn_wmma_f32_16x16x4_f32(false, a, false, a,
                                                    (short)0, acc, false, false);
    }

    // C/D layout: VGPR p, lanes 0-15 -> row p, lanes 16-31 -> row p+8; col = lane&15
    const int mbase = (lane >= 16) ? 8 : 0;
    const int ncol  = lane & 15;
#pragma unroll
    for (int p = 0; p < 8; ++p)
        sG[wave * 256 + (p + mbase) * 16 + ncol] = acc[p];
    __syncthreads();

    float g = 0.f;
#pragma unroll
    for (int w = 0; w < 8; ++w) g += sG[w * 256 + tid];
    sG[8 * 256 + tid] = g;
    __syncthreads();

    if (tid == 0) {
        const float inv_n  = 1.0f / (float)NN;
        const float inv_n2 = inv_n * inv_n;
        const float* G = sG + 8 * 256;                // [16][16]
        float total = 0.f;
        for (int l = 0; l < 2; ++l) {
            const float* S = ws + WS_S + l * 7;       // 00,01,11,12,22,23,33
            const float* T = ws + WS_T + l * 4;
            const int rb = l * 8;
            float cs[7];
            const int pa[7] = {0, 0, 1, 1, 2, 2, 3};
            const int pb[7] = {0, 1, 1, 2, 2, 3, 3};
#pragma unroll
            for (int q = 0; q < 7; ++q) {
                int a = pa[q], b = pb[q];
                float RR = G[(rb + a) * 16 + (rb + b)];
                float CC = G[(rb + 4 + a) * 16 + (rb + 4 + b)];
                cs[q] = S[q] - (RR + CC) * inv_n + T[a] * T[b] * inv_n2;
            }
            float v[7];
#pragma unroll
            for (int q = 0; q < 7; ++q)
                v[q] = sqrtf(fmaxf(cs[q] * inv_n2, 0.f) + 1e-8f);
            // v: 00,01,11,12,22,23,33
            float loss = v[1] / (sqrtf(fmaxf(v[0] * v[2], 0.f)) + 1e-10f)
                       + v[3] / (sqrtf(fmaxf(v[2] * v[4], 0.f)) + 1e-10f)
                       + v[5] / (sqrtf(fmaxf(v[4] * v[6], 0.f)) + 1e-10f);
            total += loss * 0.1f;                     // / ((NF+1)*NF/2) = /10
        }
        out[0] = total;
    }
}

// ---------------------------------------------------------------- launch
extern "C" void kernel_launch(void* const* d_in, const int* in_sizes, int n_in,
                              void* d_out, int out_size, void* d_ws, size_t ws_size,
                              hipStream_t stream) {
    const float* users         = (const float*)d_in[0];
    const float* pos_items     = (const float*)d_in[1];
    const float* users_pop     = (const float*)d_in[2];
    const float* pos_items_pop = (const float*)d_in[3];
    float* ws  = (float*)d_ws;
    float* out = (float*)d_out;

    k_rows<<<(2 * NN + 255) / 256, 256, 0, stream>>>(users, pos_items, users_pop,
                                                     pos_items_pop, ws);
    k_zero<<<(WS_ACC_COUNT + 255) / 256, 256, 0, stream>>>(ws);
    k_main<<<dim3(8, 16, 2), 256, 0, stream>>>(ws, ws, ws, ws);
    k_final<<<1, 256, 0, stream>>>(ws, out);
}